// CVaROptimizationLayer_84559316124319
// MI455X (gfx1250) — compile-verified
//
#include <hip/hip_runtime.h>
#include <hip/hip_bf16.h>
#include <math.h>

// ---------------------------------------------------------------------------
// CVaR optimization layer for MI455X (gfx1250, wave32).
//   B=64 batches, N=256 assets, S=2000 scenarios, K=100 (top-5% CVaR), 200 iters
// Pipeline:
//   1) chol_kernel : L = chol(sigma + 1e-4 I)           (64 blocks x 256 thr)
//   2) gemm_kernel : R = mu + eps @ L^T  via V_WMMA_F32_16X16X4_F32,
//                    2x2 register-blocked (32x32 macro-tile per wave)
//   3) solver_kernel: 200 projected-subgradient iterations, one block/batch
// Workspace layout (floats): R[64*2000*256] then L[64*256*256]  (~148 MB)
// ---------------------------------------------------------------------------

#define BATCH 64
#define NA    256
#define SNUM  2000
#define KSEL  100
#define BILIX 4
#define NITER 200
#define JIT   1e-4f

typedef float v2f __attribute__((ext_vector_type(2)));
typedef float v8f __attribute__((ext_vector_type(8)));

// ------------------------------- Cholesky ---------------------------------
__global__ __launch_bounds__(256) void chol_kernel(const float* __restrict__ sigma,
                                                   float* __restrict__ Lb) {
    int b = blockIdx.x;
    int tid = threadIdx.x;
    const float* A = sigma + (size_t)b * NA * NA;
    float* L = Lb + (size_t)b * NA * NA;
    for (int i = tid; i < NA * NA; i += 256) {
        float v = A[i];
        if ((i / NA) == (i % NA)) v += JIT;
        L[i] = v;
    }
    __syncthreads();
    for (int j = 0; j < NA; ++j) {
        if (tid == j) {
            float s = L[(size_t)j * NA + j];
            for (int k = 0; k < j; ++k) { float x = L[(size_t)j * NA + k]; s -= x * x; }
            L[(size_t)j * NA + j] = sqrtf(fmaxf(s, 1e-20f));
        }
        __syncthreads();
        float Ljj = L[(size_t)j * NA + j];
        if (tid > j) {
            float s = L[(size_t)tid * NA + j];
            for (int k = 0; k < j; ++k) s -= L[(size_t)tid * NA + k] * L[(size_t)j * NA + k];
            L[(size_t)tid * NA + j] = s / Ljj;
        } else if (tid < j) {
            L[(size_t)tid * NA + j] = 0.0f;   // zero upper triangle (like jnp.cholesky)
        }
        __syncthreads();
    }
}

// ---------------------- R = mu + eps @ L^T  (fp32 WMMA) --------------------
// grid: (4 m-blocks of 64 cols, 16 s-blocks of 128 rows, 64 batches)
// block: 256 threads = 8 waves arranged 4(s) x 2(m); each wave computes a
// 32x32 macro-tile = 2x2 WMMA tiles -> 4 fragment loads feed 4 WMMAs.
#define KC 64
#define LPAD 4
__global__ __launch_bounds__(256) void gemm_kernel(const float* __restrict__ eps,
                                                   const float* __restrict__ Lb,
                                                   const float* __restrict__ mu,
                                                   const int* __restrict__ lam_p,
                                                   float* __restrict__ R) {
    __shared__ __align__(16) float sA[128][KC + LPAD];  // eps panel (rows x k)
    __shared__ __align__(16) float sB[64][KC + LPAD];   // L tile    (cols x k)

    int b     = blockIdx.z;
    int m0    = blockIdx.x * 64;
    int sbase = blockIdx.y * 128;
    int tid   = threadIdx.x;
    int lane  = tid & 31;
    int wave  = tid >> 5;
    int ws    = wave & 3;    // s position (32 rows each)
    int wm    = wave >> 2;   // m position (32 cols each)

    const float* epsb  = eps + (size_t)b * SNUM * NA;
    const float* Lbase = Lb + ((size_t)b * NA + m0) * NA;   // rows m0..m0+63 of L

    v8f acc00, acc01, acc10, acc11;
    #pragma unroll
    for (int v = 0; v < 8; ++v) { acc00[v] = 0.f; acc01[v] = 0.f; acc10[v] = 0.f; acc11[v] = 0.f; }

    for (int kc = 0; kc < NA; kc += KC) {
        // cooperative load eps panel: 128 rows x KC, float4-wide (coalesced)
        for (int idx = tid; idx < 128 * (KC / 4); idx += 256) {
            int r = idx / (KC / 4), k4 = idx % (KC / 4);
            int s = sbase + r; if (s > SNUM - 1) s = SNUM - 1;
            float4 val = *(const float4*)(epsb + (size_t)s * NA + kc + k4 * 4);
            *(float4*)(&sA[r][k4 * 4]) = val;
        }
        // cooperative load L tile: 64 rows x KC, float4-wide
        for (int idx = tid; idx < 64 * (KC / 4); idx += 256) {
            int n = idx / (KC / 4), k4 = idx % (KC / 4);
            float4 val = *(const float4*)(Lbase + (size_t)n * NA + kc + k4 * 4);
            *(float4*)(&sB[n][k4 * 4]) = val;
        }
        __syncthreads();

        // fragment indexing: A[m][k], m = lane&15, k = v + 2*(lane>>4)
        //                    B[k][n] = L[col n][k], n = lane&15, same k pattern
        int koff = 2 * (lane >> 4);
        int fr   = lane & 15;
        const float* a0p = &sA[ws * 32 + fr][0];
        const float* a1p = &sA[ws * 32 + 16 + fr][0];
        const float* b0p = &sB[wm * 32 + fr][0];
        const float* b1p = &sB[wm * 32 + 16 + fr][0];
        #pragma unroll
        for (int kk = 0; kk < KC; kk += 4) {
            v2f a0, a1, b0, b1;
            a0[0] = a0p[kk + koff]; a0[1] = a0p[kk + koff + 1];
            a1[0] = a1p[kk + koff]; a1[1] = a1p[kk + koff + 1];
            b0[0] = b0p[kk + koff]; b0[1] = b0p[kk + koff + 1];
            b1[0] = b1p[kk + koff]; b1[1] = b1p[kk + koff + 1];
            acc00 = __builtin_amdgcn_wmma_f32_16x16x4_f32(false, a0, false, b0, (short)0, acc00, false, false);
            acc01 = __builtin_amdgcn_wmma_f32_16x16x4_f32(false, a0, false, b1, (short)0, acc01, false, false);
            acc10 = __builtin_amdgcn_wmma_f32_16x16x4_f32(false, a1, false, b0, (short)0, acc10, false, false);
            acc11 = __builtin_amdgcn_wmma_f32_16x16x4_f32(false, a1, false, b1, (short)0, acc11, false, false);
        }
        __syncthreads();
    }

    // epilogue: D layout -> M = v + 8*(lane>>4), Nidx = lane&15
    int lamv = lam_p[0];
    float mscale = (lamv > 0) ? (1.0f + 1.0f / fmaxf((float)lamv, 0.1f)) : 1.0f;
    int n = lane & 15;
    float mu0 = mu[(size_t)b * NA + m0 + wm * 32 + n] * mscale;
    float mu1 = mu[(size_t)b * NA + m0 + wm * 32 + 16 + n] * mscale;
    int mrow0 = 8 * (lane >> 4);
    #pragma unroll
    for (int v = 0; v < 8; ++v) {
        int s0 = sbase + ws * 32 + mrow0 + v;
        int s1 = s0 + 16;
        size_t col = (size_t)m0 + wm * 32 + n;
        if (s0 < SNUM) {
            float* r0 = R + ((size_t)b * SNUM + s0) * NA + col;
            r0[0]  = acc00[v] + mu0;
            r0[16] = acc01[v] + mu1;
        }
        if (s1 < SNUM) {
            float* r1 = R + ((size_t)b * SNUM + s1) * NA + col;
            r1[0]  = acc10[v] + mu0;
            r1[16] = acc11[v] + mu1;
        }
    }
}

// ------------------------------- Solver ------------------------------------
__device__ __forceinline__ unsigned f2key(float f) {
    unsigned u = __float_as_uint(f);
    return (u & 0x80000000u) ? ~u : (u | 0x80000000u);
}

// Euclidean projection of s_v (256 vals, = w_pre - floor) onto {x>=0, sum=mass},
// then adds floor back and stores into s_w. Uniform control flow for barriers.
__device__ void block_project(int tid, float mass, float floorv,
                              float* s_v, float* s_u, float* s_scan,
                              unsigned* s_cnt, float* s_w) {
    // rank-based descending sort (stable), O(N^2)/block = trivial for N=256
    if (tid < NA) {
        float v = s_v[tid];
        int rank = 0;
        for (int j = 0; j < NA; ++j) {
            float vj = s_v[j];
            rank += (vj > v) || (vj == v && j < tid);
        }
        s_u[rank] = v;
    }
    __syncthreads();
    if (tid < NA) s_scan[tid] = s_u[tid];
    __syncthreads();
    for (int off = 1; off < NA; off <<= 1) {   // inclusive Hillis-Steele scan
        float add = 0.0f;
        if (tid < NA && tid >= off) add = s_scan[tid - off];
        __syncthreads();
        if (tid < NA) s_scan[tid] += add;
        __syncthreads();
    }
    if (tid == 0) *s_cnt = 0u;
    __syncthreads();
    if (tid < NA) {
        float css = s_scan[tid] - mass;
        if (s_u[tid] - css / (float)(tid + 1) > 0.0f) atomicAdd(s_cnt, 1u);
    }
    __syncthreads();
    int rho = (int)*s_cnt; if (rho < 1) rho = 1;
    float theta = (s_scan[rho - 1] - mass) / (float)rho;
    if (tid < NA)
        s_w[tid] = fmaxf(s_v[tid] - theta, 0.0f) + ((tid == BILIX) ? floorv : 0.0f);
    __syncthreads();
}

__global__ __launch_bounds__(1024) void solver_kernel(const float* __restrict__ R,
                                                      const int* __restrict__ crisis_p,
                                                      float* __restrict__ out) {
    __shared__ float    s_w[NA], s_v[NA], s_u[NA], s_scan[NA], s_gsum[NA];
    __shared__ unsigned s_key[SNUM];
    __shared__ unsigned s_hist[256];
    __shared__ unsigned s_cnt, s_seld, s_above;
    __shared__ float    s_fsum;

    int b    = blockIdx.x;
    int tid  = threadIdx.x;
    int lane = tid & 31;
    int wave = tid >> 5;
    const float* Rb = R + (size_t)b * SNUM * NA;

    float floorv = 0.5f * (float)crisis_p[0];
    float mass   = 1.0f - floorv;

    // w0 = proj(1/N)
    if (tid < NA) s_v[tid] = (1.0f / (float)NA) - ((tid == BILIX) ? floorv : 0.0f);
    __syncthreads();
    block_project(tid, mass, floorv, s_v, s_u, s_scan, &s_cnt, s_w);

    for (int it = 0; it < NITER; ++it) {
        // ---- pass 1: losses[s] = -R[s,:].w  -> sortable keys (wave per row) ----
        for (int s = wave; s < SNUM; s += 32) {
            const float* row = Rb + (size_t)s * NA;
            float p = 0.0f;
            #pragma unroll
            for (int j = 0; j < 8; ++j) {
                int n = lane + 32 * j;
                p += row[n] * s_w[n];
            }
            #pragma unroll
            for (int off = 16; off > 0; off >>= 1) p += __shfl_xor(p, off, 32);
            if (lane == 0) s_key[s] = f2key(-p);
        }
        __syncthreads();

        // ---- exact K-th largest via 8-bit-digit radix select ----
        unsigned prefix = 0u, kneed = KSEL;
        for (int pass = 0; pass < 4; ++pass) {
            int shift = 24 - 8 * pass;
            unsigned himask = (pass == 0) ? 0u : (0xFFFFFFFFu << (shift + 8));
            if (tid < 256) s_hist[tid] = 0u;
            __syncthreads();
            for (int s = tid; s < SNUM; s += 1024) {
                unsigned k = s_key[s];
                if ((k & himask) == prefix) atomicAdd(&s_hist[(k >> shift) & 255u], 1u);
            }
            __syncthreads();
            if (tid == 0) {
                unsigned acc = 0u; int d = 255;
                for (; d > 0; --d) {
                    unsigned h = s_hist[d];
                    if (acc + h >= kneed) break;
                    acc += h;
                }
                s_seld = (unsigned)d;
                s_above = acc;
            }
            __syncthreads();
            prefix |= (s_seld << shift);
            kneed  -= s_above;
            __syncthreads();
        }
        unsigned kth = prefix;

        // ---- count(mask) for denom = max(cnt, K) ----
        if (tid == 0) s_cnt = 0u;
        __syncthreads();
        {
            unsigned c = 0u;
            for (int s = tid; s < SNUM; s += 1024) c += (s_key[s] >= kth);
            atomicAdd(&s_cnt, c);
        }
        __syncthreads();
        float denom = fmaxf((float)s_cnt, (float)KSEL);

        // ---- pass 2: gsum[n] = sum_{mask} R[s,n]  (mask is wave-uniform) ----
        if (tid < NA) s_gsum[tid] = 0.0f;
        __syncthreads();
        {
            float gp[8];
            #pragma unroll
            for (int j = 0; j < 8; ++j) gp[j] = 0.0f;
            for (int s = wave; s < SNUM; s += 32) {
                if (s_key[s] >= kth) {
                    const float* row = Rb + (size_t)s * NA;
                    #pragma unroll
                    for (int j = 0; j < 8; ++j) gp[j] += row[lane + 32 * j];
                }
            }
            #pragma unroll
            for (int j = 0; j < 8; ++j) atomicAdd(&s_gsum[lane + 32 * j], gp[j]);
        }
        __syncthreads();

        // ---- update + projection:  w = proj(w - lr*g),  g = -gsum/denom ----
        float lr = 0.5f / sqrtf((float)it + 1.0f);
        if (tid < NA) {
            float wpre = s_w[tid] + lr * s_gsum[tid] / denom;
            s_v[tid] = wpre - ((tid == BILIX) ? floorv : 0.0f);
        }
        __syncthreads();
        block_project(tid, mass, floorv, s_v, s_u, s_scan, &s_cnt, s_w);
    }

    // ---- final clamp + renormalize ----
    if (tid == 0) s_fsum = 0.0f;
    __syncthreads();
    float wv = 0.0f;
    if (tid < NA) {
        wv = fmaxf(s_w[tid], 0.0f);
        atomicAdd(&s_fsum, wv);
    }
    __syncthreads();
    if (tid < NA) out[(size_t)b * NA + tid] = wv / (s_fsum + 1e-8f);
}

// ------------------------------ launcher -----------------------------------
extern "C" void kernel_launch(void* const* d_in, const int* in_sizes, int n_in,
                              void* d_out, int out_size, void* d_ws, size_t ws_size,
                              hipStream_t stream) {
    const float* mu     = (const float*)d_in[0];
    const float* sigma  = (const float*)d_in[1];
    const float* eps    = (const float*)d_in[2];
    const int*   crisis = (const int*)d_in[3];
    const int*   lam    = (const int*)d_in[4];
    float*       out    = (float*)d_out;

    float* ws   = (float*)d_ws;
    float* Rbuf = ws;                                      // 64*2000*256 floats
    float* Lbuf = ws + (size_t)BATCH * SNUM * NA;          // 64*256*256 floats

    chol_kernel<<<BATCH, 256, 0, stream>>>(sigma, Lbuf);
    gemm_kernel<<<dim3(NA / 64, (SNUM + 127) / 128, BATCH), 256, 0, stream>>>(
        eps, Lbuf, mu, lam, Rbuf);
    solver_kernel<<<BATCH, 1024, 0, stream>>>(Rbuf, crisis, out);
}